// Net_19258633356195
// MI455X (gfx1250) — compile-verified
//
#include <hip/hip_runtime.h>

typedef __attribute__((ext_vector_type(2))) float v2f;
typedef __attribute__((ext_vector_type(4))) float v4f;
typedef __attribute__((ext_vector_type(8))) float v8f;

#define WAVES_PER_WG 8

// Broadcast lane `l`'s value to all lanes via v_readlane (VALU -> SGPR, no DS).
__device__ __forceinline__ float lane_bcast(float x, int l) {
  return __uint_as_float(__builtin_amdgcn_readlane(__float_as_uint(x), l));
}

// ---------------------------------------------------------------------------
// Zero the 64-word max-accumulator in workspace (relu outputs are >= 0, so
// 0u == 0.0f is the identity for the max reduction).
// ---------------------------------------------------------------------------
__global__ void zero64_kernel(unsigned int* __restrict__ p) {
  p[threadIdx.x] = 0u;
}

// Row pointer for tile `rt`: row 0 = self, rows 1..32 = neighbors; rows >32
// clamp to 32 (duplicate last neighbor; harmless under the max-pool).
__device__ __forceinline__ const float* row_ptr(const float* __restrict__ x_self,
                                                const float* __restrict__ x_nb,
                                                int n, int rt, int lq) {
  int row = rt * 16 + lq;
  if (row > 32) row = 32;
  return (row == 0) ? (x_self + (size_t)n * 64)
                    : (x_nb + ((size_t)n * 32 + (row - 1)) * 64);
}

// Load one 16x64 A-tile as WMMA fragments: lane(lq,half), vgpr j ->
// X[row = lq][d = kk*4 + half*2 + j].
__device__ __forceinline__ void load_tile(v2f (&A)[16], const float* __restrict__ rp,
                                          int half) {
#pragma unroll
  for (int kk = 0; kk < 16; ++kk)
    A[kk] = *(const v2f*)(rp + kk * 4 + half * 2);
}

// 16x64 x 64x64 tile GEMM (64 WMMAs) + bias + relu + 16-row max into colmax.
__device__ __forceinline__ void mm_tile(const v2f (&A)[16], const v2f (&Bf)[16][4],
                                        const float (&bb)[4], float (&colmax)[4]) {
  v8f cacc[4] = {};
#pragma unroll
  for (int kk = 0; kk < 16; ++kk) {
#pragma unroll
    for (int nt = 0; nt < 4; ++nt) {
      cacc[nt] = __builtin_amdgcn_wmma_f32_16x16x4_f32(
          false, A[kk], false, Bf[kk][nt], (short)0, cacc[nt], false, false);
    }
  }
  // C layout: vgpr i, lane -> row = 8*half + i, col = nt*16 + lq.
#pragma unroll
  for (int nt = 0; nt < 4; ++nt) {
    float m = 0.f;
#pragma unroll
    for (int i = 0; i < 8; ++i)
      m = fmaxf(m, fmaxf(cacc[nt][i] + bb[nt], 0.f));
    m = fmaxf(m, __shfl_xor(m, 16, 32));  // combine row halves
    colmax[nt] = fmaxf(colmax[nt], m);
  }
}

// ---------------------------------------------------------------------------
// Fused stage 1 + stage 2 row computation. One wave == one node (grid-stride).
//   aggr   = max over 33 rows of relu(X @ W1^T + b1)      (WMMA f32 16x16x4)
//   embs1  = l2norm(relu(aggr @ emb1W^T))                 (VALU, 2 outs/lane)
//   r      = relu(embs1 @ W2a^T + b2)                     (VALU, 2 outs/lane)
//   global elementwise max of r across nodes -> out_max   (uint atomicMax)
//
// Software-pipelined: tile 0 of node n is loaded during node n-1's tile-2
// WMMAs + epilogue; tiles 1,2 load while tile 0 is consumed. Keeps 32-48
// loads (8-12 KB) in flight per wave. Uncapped VGPRs: the working set
// (Bf 128 + 3x32 A + 32 C) must not spill; MSB SALU traffic is cheaper.
// ---------------------------------------------------------------------------
__global__ void __launch_bounds__(256, 1)
sage_stage1_kernel(const float* __restrict__ x_self,
                   const float* __restrict__ x_nb,
                   const float* __restrict__ W1,
                   const float* __restrict__ b1,
                   const float* __restrict__ emb1W,
                   const float* __restrict__ W2a,
                   const float* __restrict__ b2,
                   unsigned int* __restrict__ out_max,
                   int N, int total_waves) {
  const int tid   = threadIdx.x;
  const int lane  = tid & 31;
  const int wv    = tid >> 5;
  const int lq    = lane & 15;   // N / M position within fragment
  const int half  = lane >> 4;   // K-half selector
  const int gwave = blockIdx.x * WAVES_PER_WG + wv;

  // --- Preload B fragments of W1 (node-invariant, lives in VGPRs) ---------
  // B frag (nt, kk): lane(lq,half), vgpr j -> W1[o = nt*16+lq][d = kk*4 + half*2 + j]
  v2f Bf[16][4];
#pragma unroll
  for (int kk = 0; kk < 16; ++kk) {
#pragma unroll
    for (int nt = 0; nt < 4; ++nt) {
      Bf[kk][nt] = *(const v2f*)(W1 + (nt * 16 + lq) * 64 + kk * 4 + half * 2);
    }
  }
  float bb[4];
#pragma unroll
  for (int nt = 0; nt < 4; ++nt) bb[nt] = b1[nt * 16 + lq];
  const float b2a = b2[lane];
  const float b2b = b2[lane + 32];

  float r1max = 0.f, r2max = 0.f;  // running stage-2 max (relu => >= 0)

  v2f A0[16], A1[16], A2[16];
  if (gwave < N) load_tile(A0, row_ptr(x_self, x_nb, gwave, 0, lq), half);

  for (int n = gwave; n < N; n += total_waves) {
    const int nn = n + total_waves;
    // Prefetch the next node's data one full node ahead of its real loads.
    if (nn < N) {
      const float* pn = x_nb + ((size_t)nn * 32 + lane) * 64;
      __builtin_prefetch(pn, 0, 1);
      __builtin_prefetch(pn + 32, 0, 1);
      __builtin_prefetch(x_self + (size_t)nn * 64 + lane * 2, 0, 1);
    }

    // Issue tiles 1 and 2 now; consume tile 0 (in flight since last iter).
    load_tile(A1, row_ptr(x_self, x_nb, n, 1, lq), half);
    load_tile(A2, row_ptr(x_self, x_nb, n, 2, lq), half);

    float colmax[4] = {0.f, 0.f, 0.f, 0.f};
    mm_tile(A0, Bf, bb, colmax);
    mm_tile(A1, Bf, bb, colmax);
    // Next node's tile 0 flies through tile-2 WMMAs and the whole epilogue.
    if (nn < N) load_tile(A0, row_ptr(x_self, x_nb, nn, 0, lq), half);
    mm_tile(A2, Bf, bb, colmax);
    // colmax[nt] (all lanes): aggr value for column nt*16 + lq.

    // ---- embs1 = l2norm(relu(aggr @ emb1W^T)); outputs o=lane, o=lane+32 ---
    float e1 = 0.f, e2 = 0.f;
    const float* wr1 = emb1W + lane * 64;
    const float* wr2 = emb1W + (lane + 32) * 64;
#pragma unroll
    for (int nt = 0; nt < 4; ++nt) {
#pragma unroll
      for (int dc = 0; dc < 4; ++dc) {
        v4f wa = *(const v4f*)(wr1 + nt * 16 + dc * 4);
        v4f wb = *(const v4f*)(wr2 + nt * 16 + dc * 4);
#pragma unroll
        for (int j = 0; j < 4; ++j) {
          float av = lane_bcast(colmax[nt], dc * 4 + j);  // broadcast aggr[d]
          e1 = fmaf(av, wa[j], e1);
          e2 = fmaf(av, wb[j], e2);
        }
      }
    }
    e1 = fmaxf(e1, 0.f);
    e2 = fmaxf(e2, 0.f);
    float ss = e1 * e1 + e2 * e2;
#pragma unroll
    for (int mk = 16; mk >= 1; mk >>= 1) ss += __shfl_xor(ss, mk, 32);
    float nrm = sqrtf(ss);
    float inv = (nrm > 0.f) ? (1.f / nrm) : 1.f;
    e1 *= inv;
    e2 *= inv;

    // ---- r = relu(embs1 @ W2a^T + b2); running global max -----------------
    float r1 = b2a, r2 = b2b;
    const float* ar1 = W2a + lane * 64;
    const float* ar2 = W2a + (lane + 32) * 64;
#pragma unroll
    for (int hh = 0; hh < 2; ++hh) {
      float src = hh ? e2 : e1;
#pragma unroll
      for (int dc = 0; dc < 8; ++dc) {
        v4f wa = *(const v4f*)(ar1 + hh * 32 + dc * 4);
        v4f wb = *(const v4f*)(ar2 + hh * 32 + dc * 4);
#pragma unroll
        for (int j = 0; j < 4; ++j) {
          float av = lane_bcast(src, dc * 4 + j);
          r1 = fmaf(av, wa[j], r1);
          r2 = fmaf(av, wb[j], r2);
        }
      }
    }
    r1max = fmaxf(r1max, fmaxf(r1, 0.f));
    r2max = fmaxf(r2max, fmaxf(r2, 0.f));
  }

  // ---- workgroup reduce, then one atomicMax per element per workgroup -----
  __shared__ float red[WAVES_PER_WG * 64];
  red[wv * 64 + lane]      = r1max;
  red[wv * 64 + 32 + lane] = r2max;
  __syncthreads();
  if (tid < 64) {
    float m = red[tid];
#pragma unroll
    for (int w = 1; w < WAVES_PER_WG; ++w) m = fmaxf(m, red[w * 64 + tid]);
    // values are >= 0, so uint compare == float compare on bit patterns
    atomicMax(out_max + tid, __float_as_uint(m));
  }
}

// ---------------------------------------------------------------------------
// Finalize: emb2 = l2norm(relu(aggr2 @ emb2W^T)); h = relu(emb2 @ W1r^T + b1r);
// out = h @ W2r^T + b2r.  One wave, two outputs per lane.
// ---------------------------------------------------------------------------
__global__ void sage_finalize_kernel(const unsigned int* __restrict__ ws,
                                     const float* __restrict__ emb2W,
                                     const float* __restrict__ rW1,
                                     const float* __restrict__ rb1,
                                     const float* __restrict__ rW2,
                                     const float* __restrict__ rb2,
                                     float* __restrict__ out) {
  const int l = threadIdx.x;  // 32 threads
  float a1 = __uint_as_float(ws[l]);
  float a2 = __uint_as_float(ws[32 + l]);

  float g1 = 0.f, g2 = 0.f;
  const float* er1 = emb2W + l * 64;
  const float* er2 = emb2W + (l + 32) * 64;
#pragma unroll
  for (int hh = 0; hh < 2; ++hh) {
    float src = hh ? a2 : a1;
#pragma unroll
    for (int d = 0; d < 32; ++d) {
      float av = lane_bcast(src, d);
      g1 = fmaf(av, er1[hh * 32 + d], g1);
      g2 = fmaf(av, er2[hh * 32 + d], g2);
    }
  }
  g1 = fmaxf(g1, 0.f);
  g2 = fmaxf(g2, 0.f);
  float ss = g1 * g1 + g2 * g2;
#pragma unroll
  for (int mk = 16; mk >= 1; mk >>= 1) ss += __shfl_xor(ss, mk, 32);
  float nrm = sqrtf(ss);
  float inv = (nrm > 0.f) ? (1.f / nrm) : 1.f;
  g1 *= inv;
  g2 *= inv;

  float h1 = rb1[l], h2 = rb1[l + 32];
  const float* hr1 = rW1 + l * 64;
  const float* hr2 = rW1 + (l + 32) * 64;
#pragma unroll
  for (int hh = 0; hh < 2; ++hh) {
    float src = hh ? g2 : g1;
#pragma unroll
    for (int d = 0; d < 32; ++d) {
      float av = lane_bcast(src, d);
      h1 = fmaf(av, hr1[hh * 32 + d], h1);
      h2 = fmaf(av, hr2[hh * 32 + d], h2);
    }
  }
  h1 = fmaxf(h1, 0.f);
  h2 = fmaxf(h2, 0.f);
  float part = h1 * rW2[l] + h2 * rW2[l + 32];
#pragma unroll
  for (int mk = 16; mk >= 1; mk >>= 1) part += __shfl_xor(part, mk, 32);
  if (l == 0) out[0] = part + rb2[0];
}

extern "C" void kernel_launch(void* const* d_in, const int* in_sizes, int n_in,
                              void* d_out, int out_size, void* d_ws, size_t ws_size,
                              hipStream_t stream) {
  const float* x_self = (const float*)d_in[0];
  const float* x_nb   = (const float*)d_in[1];
  const float* W1     = (const float*)d_in[2];   // aggr1_W [64,64]
  const float* b1v    = (const float*)d_in[3];   // aggr1_b [64]
  const float* emb1W  = (const float*)d_in[4];   // emb1_W  [64,64]
  const float* W2a    = (const float*)d_in[5];   // aggr2_W [64,64]
  const float* b2v    = (const float*)d_in[6];   // aggr2_b [64]
  const float* emb2W  = (const float*)d_in[7];   // emb2_W  [64,64]
  const float* rW1    = (const float*)d_in[8];   // reg_W1  [64,64]
  const float* rb1    = (const float*)d_in[9];   // reg_b1  [64]
  const float* rW2    = (const float*)d_in[10];  // reg_W2  [1,64]
  const float* rb2    = (const float*)d_in[11];  // reg_b2  [1]
  float* out          = (float*)d_out;
  unsigned int* ws    = (unsigned int*)d_ws;     // 64-word max accumulator

  const int N = in_sizes[0] / 64;  // nodes

  int nwg = (N + WAVES_PER_WG - 1) / WAVES_PER_WG;
  if (nwg > 2560) nwg = 2560;
  if (nwg < 1) nwg = 1;
  const int total_waves = nwg * WAVES_PER_WG;

  zero64_kernel<<<1, 64, 0, stream>>>(ws);
  sage_stage1_kernel<<<nwg, 256, 0, stream>>>(x_self, x_nb, W1, b1v, emb1W,
                                              W2a, b2v, ws, N, total_waves);
  sage_finalize_kernel<<<1, 32, 0, stream>>>(ws, emb2W, rW1, rb1, rW2, rb2, out);
}